// WaveletDecomposer_31860067401740
// MI455X (gfx1250) — compile-verified
//
#include <hip/hip_runtime.h>

// ---------------------------------------------------------------------------
// db4 (8-tap) wavelet bank, 4-level analysis + two partial reconstructions,
// expressed as banded-Toeplitz GEMMs on V_WMMA_F32_16X16X4_F32 (wave32).
// One wave computes a 16-position x 16-row output tile.
//   D(16 pos x 16 rows) = sum_c  A_c(16x4, constant Toeplitz slice)
//                                x B_c(4x16, signal samples)
// A-operands are compile-time tables indexed by [chunk][lane]; interior tiles
// use b64 vector loads / b128 vector stores (all buffers have even strides).
// ---------------------------------------------------------------------------

typedef __attribute__((ext_vector_type(2))) float v2f;
typedef __attribute__((ext_vector_type(4))) float v4f;
typedef __attribute__((ext_vector_type(8))) float v8f;

// db4 scaling filter h[0..7]
constexpr float Hc[8] = {
    -0.010597401784997278f,  0.032883011666982945f,
     0.030841381835986965f, -0.18703481171888114f,
    -0.02798376941698385f,   0.6308807679295904f,
     0.7148465705525415f,    0.23037781330885523f };

constexpr float hz(int j)  { return (j >= 0 && j < 8) ? Hc[j] : 0.0f; }
// DEC_HI[j] = -(-1)^j h[7-j]
constexpr float gz(int j)  {
    return (j >= 0 && j < 8) ? ((j & 1) ? Hc[7 - j] : -Hc[7 - j]) : 0.0f;
}

// ---- analysis A-tables: kk relative to T0 = 2*t0 - 6;  j = 2m + 7 - kk ----
struct DwtTab { float lo[10][32][2]; float hi[10][32][2]; };
constexpr DwtTab make_dwt_tab() {
    DwtTab t{};
    for (int c = 0; c < 10; ++c)
        for (int l = 0; l < 32; ++l)
            for (int v = 0; v < 2; ++v) {
                const int m  = l & 15;
                const int kk = 4 * c + 2 * (l >> 4) + v;
                const int j  = 2 * m + 7 - kk;
                t.lo[c][l][v] = hz(j);
                t.hi[c][l][v] = gz(j);
            }
    return t;
}
__device__ __constant__ DwtTab c_dwt = make_dwt_tab();

// ---- synthesis A-tables: u = U0 + kk with U0 = t0/2 - 2 (even) ------------
//   out[t] += rl[j]*ca[u] + rh[j]*cd[u],  j = t + 6 - 2u = m + 10 - 2kk
//   rl[j] = h[7-j] -> h[2kk - m - 3];  rh[j] = (-1)^j h[j] -> sgn(m)*h[j]
struct IdwtTab { float lo[4][32][2]; float hi[4][32][2]; };
constexpr IdwtTab make_idwt_tab() {
    IdwtTab t{};
    for (int c = 0; c < 4; ++c)
        for (int l = 0; l < 32; ++l)
            for (int v = 0; v < 2; ++v) {
                const int m  = l & 15;
                const int kk = 4 * c + 2 * (l >> 4) + v;
                const int j  = m + 10 - 2 * kk;
                t.lo[c][l][v] = hz(7 - j);
                t.hi[c][l][v] = (m & 1) ? -hz(j) : hz(j);
            }
    return t;
}
__device__ __constant__ IdwtTab c_idwt = make_idwt_tab();

// half-sample symmetric reflection (pywt 'symmetric'); single reflection is
// enough since the filter span (38) << smallest signal length (319).
__device__ __forceinline__ int symref(int s, int n) {
    if (s < 0)  s = -1 - s;
    if (s >= n) s = 2 * n - 1 - s;
    return s;
}

// ---------------------------------------------------------------------------
// Analysis: (ca,cd)[p] = sum_j {h,g}[j] * a_in[sym(2p+1-j)]
// ---------------------------------------------------------------------------
__global__ void __launch_bounds__(32)
dwt16(const float* __restrict__ a_in, int n_in, int s_in,
      float* __restrict__ ca, float* __restrict__ cd, int n_out, int s_out)
{
    const int lane = threadIdx.x;
    const int half = lane >> 4;          // 0: K={0,1}, 1: K={2,3} within chunk
    const int m    = lane & 15;          // A row / B column / D column
    const int t0   = blockIdx.x << 4;    // output-position tile base
    const int row  = (blockIdx.y << 4) + m;
    const long long rb = (long long)row * s_in;
    const int T0   = 2 * t0 - 6;         // leftmost input index of tile

    const bool fastload = (T0 >= 0) && (T0 + 40 <= n_in);   // uniform

    v8f acc_lo = {};
    v8f acc_hi = {};
    if (fastload) {
#pragma unroll
        for (int c = 0; c < 10; ++c) {
            const int kk0 = 4 * c + 2 * half;
            const v2f alo = *(const v2f*)&c_dwt.lo[c][lane][0];
            const v2f ahi = *(const v2f*)&c_dwt.hi[c][lane][0];
            const v2f b   = *(const v2f*)(a_in + rb + T0 + kk0); // 8B aligned
            acc_lo = __builtin_amdgcn_wmma_f32_16x16x4_f32(
                         false, alo, false, b, (short)0, acc_lo, false, false);
            acc_hi = __builtin_amdgcn_wmma_f32_16x16x4_f32(
                         false, ahi, false, b, (short)0, acc_hi, false, false);
        }
    } else {
#pragma unroll
        for (int c = 0; c < 10; ++c) {
            const int kk0 = 4 * c + 2 * half;
            const v2f alo = *(const v2f*)&c_dwt.lo[c][lane][0];
            const v2f ahi = *(const v2f*)&c_dwt.hi[c][lane][0];
            v2f b;
            b.x = a_in[rb + symref(T0 + kk0,     n_in)];
            b.y = a_in[rb + symref(T0 + kk0 + 1, n_in)];
            acc_lo = __builtin_amdgcn_wmma_f32_16x16x4_f32(
                         false, alo, false, b, (short)0, acc_lo, false, false);
            acc_hi = __builtin_amdgcn_wmma_f32_16x16x4_f32(
                         false, ahi, false, b, (short)0, acc_hi, false, false);
        }
    }

    // D layout: VGPR q holds position t0 + q + 8*half for this lane's row.
    const long long ob = (long long)row * s_out + t0 + 8 * half;
    if (t0 + 16 <= n_out) {              // uniform fast store, 16B aligned
        *(v4f*)(ca + ob)     = __builtin_shufflevector(acc_lo, acc_lo, 0, 1, 2, 3);
        *(v4f*)(ca + ob + 4) = __builtin_shufflevector(acc_lo, acc_lo, 4, 5, 6, 7);
        *(v4f*)(cd + ob)     = __builtin_shufflevector(acc_hi, acc_hi, 0, 1, 2, 3);
        *(v4f*)(cd + ob + 4) = __builtin_shufflevector(acc_hi, acc_hi, 4, 5, 6, 7);
    } else {
#pragma unroll
        for (int q = 0; q < 8; ++q) {
            const int p = t0 + q + 8 * half;
            if (p < n_out) {
                ca[ob + q] = acc_lo[q];
                cd[ob + q] = acc_hi[q];
            }
        }
    }
}

// ---------------------------------------------------------------------------
// Synthesis: out[t] = sum_j rl[j]*ca[(t+6-j)/2] + rh[j]*cd[(t+6-j)/2]
// (even t+6-j only, zero-extended coefficients). Either ca or cd may be null.
// n_out implements the waverec "trim last sample" rule.
// ---------------------------------------------------------------------------
__global__ void __launch_bounds__(32)
idwt16(const float* __restrict__ ca, const float* __restrict__ cd,
       int n_c, int s_c, float* __restrict__ out, int n_out, int s_out)
{
    const int lane = threadIdx.x;
    const int half = lane >> 4;
    const int m    = lane & 15;
    const int t0   = blockIdx.x << 4;
    const int row  = (blockIdx.y << 4) + m;
    const long long rb = (long long)row * s_c;
    const int U0   = (t0 >> 1) - 2;      // even; tables absorb the shift

    const bool fastload = (U0 >= 0) && (U0 + 16 <= n_c);    // uniform

    v8f acc = {};
    if (fastload) {
#pragma unroll
        for (int c = 0; c < 4; ++c) {
            const int kk0 = 4 * c + 2 * half;
            if (ca) {
                const v2f a = *(const v2f*)&c_idwt.lo[c][lane][0];
                const v2f b = *(const v2f*)(ca + rb + U0 + kk0); // 8B aligned
                acc = __builtin_amdgcn_wmma_f32_16x16x4_f32(
                          false, a, false, b, (short)0, acc, false, false);
            }
            if (cd) {
                const v2f a = *(const v2f*)&c_idwt.hi[c][lane][0];
                const v2f b = *(const v2f*)(cd + rb + U0 + kk0);
                acc = __builtin_amdgcn_wmma_f32_16x16x4_f32(
                          false, a, false, b, (short)0, acc, false, false);
            }
        }
    } else {
#pragma unroll
        for (int c = 0; c < 4; ++c) {
            const int kk0 = 4 * c + 2 * half;
            const int u0  = U0 + kk0;
            const int u1  = u0 + 1;
            // clamped unconditional loads + value select (no EXEC branches)
            const int uc0 = min(max(u0, 0), n_c - 1);
            const int uc1 = min(max(u1, 0), n_c - 1);
            const bool v0 = (u0 >= 0) && (u0 < n_c);
            const bool v1 = (u1 >= 0) && (u1 < n_c);
            if (ca) {
                const v2f a = *(const v2f*)&c_idwt.lo[c][lane][0];
                v2f b;
                b.x = v0 ? ca[rb + uc0] : 0.0f;
                b.y = v1 ? ca[rb + uc1] : 0.0f;
                acc = __builtin_amdgcn_wmma_f32_16x16x4_f32(
                          false, a, false, b, (short)0, acc, false, false);
            }
            if (cd) {
                const v2f a = *(const v2f*)&c_idwt.hi[c][lane][0];
                v2f b;
                b.x = v0 ? cd[rb + uc0] : 0.0f;
                b.y = v1 ? cd[rb + uc1] : 0.0f;
                acc = __builtin_amdgcn_wmma_f32_16x16x4_f32(
                          false, a, false, b, (short)0, acc, false, false);
            }
        }
    }

    const long long ob = (long long)row * s_out + t0 + 8 * half;
    if (t0 + 16 <= n_out) {              // uniform fast store, 16B aligned
        *(v4f*)(out + ob)     = __builtin_shufflevector(acc, acc, 0, 1, 2, 3);
        *(v4f*)(out + ob + 4) = __builtin_shufflevector(acc, acc, 4, 5, 6, 7);
    } else {
#pragma unroll
        for (int q = 0; q < 8; ++q) {
            const int t = t0 + q + 8 * half;
            if (t < n_out)
                out[ob + q] = acc[q];
        }
    }
}

// ---------------------------------------------------------------------------
// Host-side pipeline (stream-ordered; 10 launches). Level lengths:
// 5000 -> 2503 -> 1255 -> 631 -> 319, stored with even strides so every
// vector access is naturally aligned. A-buffers double as reconstruction
// ping-pong once analysis is done.
// ---------------------------------------------------------------------------
extern "C" void kernel_launch(void* const* d_in, const int* in_sizes, int n_in,
                              void* d_out, int out_size, void* d_ws, size_t ws_size,
                              hipStream_t stream) {
    const float* x   = (const float*)d_in[0];
    float*       out = (float*)d_out;

    const int n0 = 5000;
    const int rows = in_sizes[0] / n0;        // 4096
    const int n1 = (n0 + 7) / 2;              // 2503
    const int n2 = (n1 + 7) / 2;              // 1255
    const int n3 = (n2 + 7) / 2;              // 631
    const int n4 = (n3 + 7) / 2;              // 319
    const int s1 = n1 + 1, s2 = n2 + 1, s3 = n3 + 1, s4 = n4 + 1; // even strides

    float* ws = (float*)d_ws;
    size_t off = 0;
    float* A1 = ws + off; off += (size_t)rows * s1;
    float* A2 = ws + off; off += (size_t)rows * s2;
    float* A3 = ws + off; off += (size_t)rows * s3;
    float* A4 = ws + off; off += (size_t)rows * s4;
    float* D1 = ws + off; off += (size_t)rows * s1;
    float* D2 = ws + off; off += (size_t)rows * s2;
    float* D3 = ws + off; off += (size_t)rows * s3;
    float* D4 = ws + off; off += (size_t)rows * s4;

    const dim3 blk(32, 1, 1);
    const int  gy = rows / 16;
    auto tiles = [](int n) { return (n + 15) / 16; };

    // ---- 4-level analysis -------------------------------------------------
    dwt16<<<dim3(tiles(n1), gy), blk, 0, stream>>>(x,  n0, n0, A1, D1, n1, s1);
    dwt16<<<dim3(tiles(n2), gy), blk, 0, stream>>>(A1, n1, s1, A2, D2, n2, s2);
    dwt16<<<dim3(tiles(n3), gy), blk, 0, stream>>>(A2, n2, s2, A3, D3, n3, s3);
    dwt16<<<dim3(tiles(n4), gy), blk, 0, stream>>>(A3, n3, s3, A4, D4, n4, s4);

    // ---- physio = waverec(a4, d4, d3, 0, 0) -------------------------------
    idwt16<<<dim3(tiles(n3), gy), blk, 0, stream>>>(A4, D4,      n4, s4, A3, n3, s3);
    idwt16<<<dim3(tiles(n2), gy), blk, 0, stream>>>(A3, D3,      n3, s3, A2, n2, s2);
    idwt16<<<dim3(tiles(n1), gy), blk, 0, stream>>>(A2, nullptr, n2, s2, A1, n1, s1);
    idwt16<<<dim3(tiles(n0), gy), blk, 0, stream>>>(A1, nullptr, n1, s1, out, n0, n0);

    // ---- identity = waverec(0, 0, 0, d2, d1) ------------------------------
    idwt16<<<dim3(tiles(n1), gy), blk, 0, stream>>>(nullptr, D2, n2, s2, A1, n1, s1);
    idwt16<<<dim3(tiles(n0), gy), blk, 0, stream>>>(A1,      D1, n1, s1,
                                                    out + (size_t)rows * n0, n0, n0);
}